// STABiLSTMLoss_43473658970670
// MI455X (gfx1250) — compile-verified
//
#include <hip/hip_runtime.h>
#include <math.h>

// ---------------------------------------------------------------------------
// Problem constants (from the reference)
// ---------------------------------------------------------------------------
constexpr int B_ = 16;
constexpr int T_ = 12;
constexpr int N_ = 10000;
constexpr int D_ = 2;
constexpr int E_ = 320000;

constexpr int SLICE      = N_ * D_;        // 20000 floats per (b,t) per tensor
constexpr int SLICE_B    = SLICE * 4;      // 80000 bytes
constexpr int VEC4_CNT   = SLICE / 4;      // 5000 float4 chunks (exact)
constexpr int SPLIT      = 4;              // workgroups per (b,t) slice
constexpr int NBLK       = B_ * T_ * SPLIT;    // 768 workgroups
constexpr int THREADS    = 256;            // 8 wave32s
constexpr int CONS_SLOTS = B_ * T_ * D_;   // 384 per-(b,t,d) conservation sums
constexpr int ACC_FLOATS = 3 + CONS_SLOTS; // [main, spatial, temporal, cons...]

// ---------------------------------------------------------------------------
// gfx1250 async global->LDS staging (ASYNCcnt path). ROCm 7.2 clang-22
// declares the builtin with typed int4 pointers:
//   (int4 addrspace(1)* gsrc, int4 addrspace(3)* ldst, i32 offset, i32 cpol)
// Guarded so the host pass (x86) falls back to plain b128 load + ds_store.
// ---------------------------------------------------------------------------
#if defined(__has_builtin)
#  if __has_builtin(__builtin_amdgcn_global_load_async_to_lds_b128)
#    define HAVE_ASYNC_LDS 1
#  else
#    define HAVE_ASYNC_LDS 0
#  endif
#  if __has_builtin(__builtin_amdgcn_s_wait_asynccnt)
#    define HAVE_WAIT_ASYNC 1
#  else
#    define HAVE_WAIT_ASYNC 0
#  endif
#else
#  define HAVE_ASYNC_LDS 0
#  define HAVE_WAIT_ASYNC 0
#endif

typedef int v4i_ __attribute__((ext_vector_type(4)));
typedef __attribute__((address_space(1))) v4i_ gas_v4i;  // global int4
typedef __attribute__((address_space(3))) v4i_ las_v4i;  // LDS int4

__device__ __forceinline__ void stage_to_lds(const float* __restrict__ g,
                                             float* __restrict__ l,
                                             int tid, int nthreads) {
#if HAVE_ASYNC_LDS
  for (int i = tid; i < VEC4_CNT; i += nthreads) {
    __builtin_amdgcn_global_load_async_to_lds_b128(
        (gas_v4i*)(g + 4 * i),   // global source
        (las_v4i*)(l + 4 * i),   // LDS destination
        /*offset=*/0, /*cpol=*/0);
  }
#else
  for (int i = tid; i < VEC4_CNT; i += nthreads) {
    ((float4*)l)[i] = ((const float4*)g)[i];
  }
#endif
}

__device__ __forceinline__ void wait_async_copies() {
#if HAVE_ASYNC_LDS
#  if HAVE_WAIT_ASYNC
  __builtin_amdgcn_s_wait_asynccnt(0);
#  else
  asm volatile("s_wait_asynccnt 0" ::: "memory");
#  endif
#endif
}

// wave32 reduction (warpSize == 32 on gfx1250)
__device__ __forceinline__ float wave_reduce(float v) {
  v += __shfl_down(v, 16, 32);
  v += __shfl_down(v, 8, 32);
  v += __shfl_down(v, 4, 32);
  v += __shfl_down(v, 2, 32);
  v += __shfl_down(v, 1, 32);
  return v;
}

// ---------------------------------------------------------------------------
// Workspace zero-init (graph-capture safe; no hipMemset)
// ---------------------------------------------------------------------------
__global__ void stab_init_kernel(float* __restrict__ acc, int n) {
  for (int i = threadIdx.x; i < n; i += blockDim.x) acc[i] = 0.0f;
}

// ---------------------------------------------------------------------------
// Main kernel: one workgroup per (b,t,split).
//   - stages pred/targ slice (2 x 80 KB) into LDS via ASYNCcnt DMA copies
//   - spatial: 4 x ds_load_b64 float2 gathers per edge
//   - main/temporal/conservation over its node sub-range (float2 vectorized)
// ---------------------------------------------------------------------------
__global__ __launch_bounds__(THREADS) void stab_loss_kernel(
    const float* __restrict__ pred, const float* __restrict__ targ,
    const int* __restrict__ ei32,   // edge_index reinterpreted as int32 pairs:
                                    // low word of int64 at element 2*k
    float* __restrict__ acc) {
  extern __shared__ float smem[];       // 2 * SLICE floats = 160000 bytes
  float2* sp2 = (float2*)smem;          // pred[b,t,:, :] as N float2 nodes
  float2* st2 = (float2*)(smem + SLICE);

  const int tid   = threadIdx.x;
  const int split = blockIdx.x % SPLIT;
  const int bt    = blockIdx.x / SPLIT;   // 0 .. B*T-1
  const int t     = bt % T_;

  const float* gp = pred + (size_t)bt * SLICE;
  const float* gt = targ + (size_t)bt * SLICE;

  // ---- stage both slices into LDS (ASYNCcnt-tracked DMA on gfx1250) ----
  stage_to_lds(gp, (float*)sp2, tid, THREADS);
  stage_to_lds(gt, (float*)st2, tid, THREADS);
  wait_async_copies();
  __syncthreads();

  // ---- spatial consistency over this block's edge range ----
  float accSpatial = 0.0f;
  const int e0 = split * (E_ / SPLIT);
  const int e1 = e0 + (E_ / SPLIT);
  for (int e = e0 + tid; e < e1; e += THREADS) {
    // int64 indices, values < 10000: low 32-bit word suffices (b32 loads)
    const int s = ei32[2 * e];
    const int d = ei32[2 * (E_ + e)];
    // 4 x ds_load_b64 gathers (8-byte aligned float2 nodes)
    const float2 ps = sp2[s];
    const float2 pd = sp2[d];
    const float2 ts = st2[s];
    const float2 td = st2[d];
    const float v0 = fabsf(ps.x - pd.x) - fabsf(ts.x - td.x);
    const float v1 = fabsf(ps.y - pd.y) - fabsf(ts.y - td.y);
    accSpatial = fmaf(v0, v0, accSpatial);
    accSpatial = fmaf(v1, v1, accSpatial);
  }

  // ---- MAE + temporal + conservation over this block's node sub-range ----
  float accMain = 0.0f, accTemp = 0.0f, accC0 = 0.0f, accC1 = 0.0f;
  const int n0 = split * (N_ / SPLIT);          // N/SPLIT = 2500 exact
  const int n1 = n0 + (N_ / SPLIT);
  const bool hasNext = (t < T_ - 1);
  const float2* gpn2 = (const float2*)(gp + SLICE);   // pred[b, t+1]
  const float2* gtn2 = (const float2*)(gt + SLICE);   // targ[b, t+1]
  for (int n = n0 + tid; n < n1; n += THREADS) {
    const float2 p = sp2[n];
    const float2 q = st2[n];
    accMain += fabsf(p.x - q.x) + fabsf(p.y - q.y);
    accC0 += p.x - q.x;                          // conservation, d = 0
    accC1 += p.y - q.y;                          // conservation, d = 1
    if (hasNext) {
      const float2 pn = gpn2[n];
      const float2 qn = gtn2[n];
      const float vx = (pn.x - p.x) - (qn.x - q.x);
      const float vy = (pn.y - p.y) - (qn.y - q.y);
      accTemp = fmaf(vx, vx, accTemp);
      accTemp = fmaf(vy, vy, accTemp);
    }
  }

  // ---- wave32 reduce, then one set of f32 atomics per wave ----
  accMain    = wave_reduce(accMain);
  accSpatial = wave_reduce(accSpatial);
  accTemp    = wave_reduce(accTemp);
  accC0      = wave_reduce(accC0);
  accC1      = wave_reduce(accC1);
  if ((tid & 31) == 0) {
    atomicAdd(&acc[0], accMain);
    atomicAdd(&acc[1], accSpatial);
    atomicAdd(&acc[2], accTemp);
    atomicAdd(&acc[3 + bt * 2 + 0], accC0);   // per-(b,t,d) node sums; squared
    atomicAdd(&acc[3 + bt * 2 + 1], accC1);   // later in finalize
  }
}

// ---------------------------------------------------------------------------
// Finalize: means, conservation square, weighted total. 5 outputs.
// ---------------------------------------------------------------------------
__global__ void stab_finalize_kernel(const float* __restrict__ acc,
                                     float* __restrict__ out) {
  if (threadIdx.x != 0 || blockIdx.x != 0) return;
  const float mainL = acc[0] / ((float)B_ * T_ * N_ * D_);        // 3.84e6
  const float spat  = acc[1] / ((float)B_ * T_ * E_ * D_);        // 1.2288e8
  const float temp  = acc[2] / ((float)B_ * (T_ - 1) * N_ * D_);  // 3.52e6
  float cs = 0.0f;
  for (int i = 0; i < CONS_SLOTS; ++i) {
    const float v = acc[3 + i];
    cs = fmaf(v, v, cs);
  }
  const float cons = cs / (float)CONS_SLOTS;
  out[0] = mainL + 0.1f * spat + 0.05f * temp + 0.02f * cons;
  out[1] = mainL;
  out[2] = spat;
  out[3] = temp;
  out[4] = cons;
}

// ---------------------------------------------------------------------------
// Entry point
// ---------------------------------------------------------------------------
extern "C" void kernel_launch(void* const* d_in, const int* in_sizes, int n_in,
                              void* d_out, int out_size, void* d_ws,
                              size_t ws_size, hipStream_t stream) {
  const float* pred = (const float*)d_in[0];          // [B,T,N,D] f32
  const float* targ = (const float*)d_in[1];          // [B,T,N,D] f32
  const int* ei32   = (const int*)d_in[2];            // [2,E] int64, low words
  float* out = (float*)d_out;                         // 5 floats
  float* acc = (float*)d_ws;                          // ACC_FLOATS floats

  stab_init_kernel<<<1, 512, 0, stream>>>(acc, ACC_FLOATS);
  stab_loss_kernel<<<dim3(NBLK), dim3(THREADS), 2 * SLICE_B, stream>>>(
      pred, targ, ei32, acc);
  stab_finalize_kernel<<<1, 1, 0, stream>>>(acc, out);
}